// InteractionNetwork_48636209660568
// MI455X (gfx1250) — compile-verified
//
#include <hip/hip_runtime.h>

// ---------------------------------------------------------------------------
// Interaction Network on MI455X (gfx1250), fp32 end-to-end via
// V_WMMA_F32_16X16X4_F32 (wave32, 16x16 tiles, K=4/instr).
//
//   k_prep      : zero-pad + pair-interleave B-operands (objects, weights)
//   k_marshal   : senders/receivers = S^T X, Rcv^T X  -> Bm [B,R,16]
//   k_relmlp    : 13->150->150->150->100 MLP (ReLU all) -> Ep (pair-interleaved)
//   k_aggregate : Rcv @ E -> effects, concat objects -> C [B*O,112]
//   k_objmlp    : 106->100->2 -> d_out [4000,2]
//
// Pair-interleaved B layout: Wp[(k>>1)*(2*Np) + n*2 + (k&1)] so the (k,k+1)
// pair needed by a lane for one WMMA B operand is a single aligned float2.
// All padding is zero-filled -> no predication (no exec-mask branches) in any
// K-loop; WMMA always executes with EXEC all-ones.
// ---------------------------------------------------------------------------

typedef __attribute__((ext_vector_type(2))) float v2f;
typedef __attribute__((ext_vector_type(8))) float v8f;

#define WMMA_F32(a, b, c) \
  __builtin_amdgcn_wmma_f32_16x16x4_f32(false, (a), false, (b), (short)0, (c), false, false)

static constexpr int B_ = 4, O_ = 1000, R_ = 20000;
static constexpr int NF = 6, ED = 100, RH = 150, OH = 100;
static constexpr int BMW = 16;   // marshalled row width (13 -> 16)
static constexpr int EW  = 112;  // padded E width     (100 -> 112)
static constexpr int CW  = 112;  // padded concat width (106 -> 112)

// workspace layout (floats)
static constexpr size_t OFF_BM   = 0;                               // 4*20000*16
static constexpr size_t OFF_EP   = OFF_BM + (size_t)B_ * R_ * BMW;  // 40000*224
static constexpr size_t OFF_C    = OFF_EP + (size_t)(B_ * R_ / 2) * (2 * EW);
static constexpr size_t OFF_OP   = OFF_C  + (size_t)B_ * O_ * CW;   // 4*500*32
static constexpr size_t OFF_W1   = OFF_OP + (size_t)B_ * 500 * 32;  // 8*320
static constexpr size_t OFF_W2   = OFF_W1 + 8 * 320;                // 80*320
static constexpr size_t OFF_W3   = OFF_W2 + 80 * 320;               // 80*320
static constexpr size_t OFF_W4   = OFF_W3 + 80 * 320;               // 80*224
static constexpr size_t OFF_OW1  = OFF_W4 + 80 * 224;               // 56*224
static constexpr size_t OFF_OW2  = OFF_OW1 + 56 * 224;              // 56*32
static constexpr size_t OFF_BIAS = OFF_OW2 + 56 * 32;               // 720
// bias sub-offsets inside bp
static constexpr int BO_B1 = 0, BO_B2 = 160, BO_B3 = 320, BO_B4 = 480,
                     BO_OB1 = 592, BO_OB2 = 704, BIAS_TOT = 720;

// ---------------------------------------------------------------------------
// Prep: pad + pair-interleave. Single writer per element, deterministic.
// ---------------------------------------------------------------------------
__device__ inline void pad_pairs(const float* __restrict__ W, float* __restrict__ Wp,
                                 int K, int N, int Kp, int Np, int tid, int nt) {
  const int total = Kp * Np;                      // == number of Wp elements
  for (int i = tid; i < total; i += nt) {
    const int rowp = i / (2 * Np);                // k >> 1
    const int rem  = i % (2 * Np);
    const int n    = rem >> 1;
    const int k    = rowp * 2 + (rem & 1);
    Wp[i] = (k < K && n < N) ? W[(size_t)k * N + n] : 0.f;
  }
}

__global__ __launch_bounds__(256) void k_prep(
    const float* __restrict__ objects,
    const float* __restrict__ rw1, const float* __restrict__ rb1,
    const float* __restrict__ rw2, const float* __restrict__ rb2,
    const float* __restrict__ rw3, const float* __restrict__ rb3,
    const float* __restrict__ rw4, const float* __restrict__ rb4,
    const float* __restrict__ ow1, const float* __restrict__ ob1,
    const float* __restrict__ ow2, const float* __restrict__ ob2,
    float* __restrict__ ws)
{
  const int tid = blockIdx.x * blockDim.x + threadIdx.x;
  const int nt  = gridDim.x * blockDim.x;

  // objects, padded 6->16 and pair-interleaved over k: Op[b][k>>1][f*2+(k&1)]
  float* Op = ws + OFF_OP;
  for (int i = tid; i < B_ * 500 * 32; i += nt) {
    const int c  = i & 31;
    const int kk = (i >> 5) % 500;
    const int b  = i / (500 * 32);
    const int k  = kk * 2 + (c & 1);
    const int f  = c >> 1;
    Op[i] = (f < NF) ? objects[((size_t)b * O_ + k) * NF + f] : 0.f;
  }

  pad_pairs(rw1, ws + OFF_W1,  13, RH, 16,  160, tid, nt);
  pad_pairs(rw2, ws + OFF_W2,  RH, RH, 160, 160, tid, nt);
  pad_pairs(rw3, ws + OFF_W3,  RH, RH, 160, 160, tid, nt);
  pad_pairs(rw4, ws + OFF_W4,  RH, ED, 160, 112, tid, nt);
  pad_pairs(ow1, ws + OFF_OW1, NF + ED, OH, 112, 112, tid, nt);
  pad_pairs(ow2, ws + OFF_OW2, OH, 2,   112, 16,  tid, nt);

  float* bp = ws + OFF_BIAS;
  for (int i = tid; i < BIAS_TOT; i += nt) {
    float v = 0.f;
    if      (i < BO_B2)  { int j = i - BO_B1;  if (j < RH) v = rb1[j]; }
    else if (i < BO_B3)  { int j = i - BO_B2;  if (j < RH) v = rb2[j]; }
    else if (i < BO_B4)  { int j = i - BO_B3;  if (j < RH) v = rb3[j]; }
    else if (i < BO_OB1) { int j = i - BO_B4;  if (j < ED) v = rb4[j]; }
    else if (i < BO_OB2) { int j = i - BO_OB1; if (j < OH) v = ob1[j]; }
    else                 { int j = i - BO_OB2; if (j < 2)  v = ob2[j]; }
    bp[i] = v;
  }
}

// ---------------------------------------------------------------------------
// Kernel 1: marshalling. A[r,k] = S[b,k,r] (contiguous in r), B = padded
// objects (one float2 per lane per k-step). Streams 640 MB once, NT-hinted.
// ---------------------------------------------------------------------------
__global__ __launch_bounds__(256) void k_marshal(
    const float* __restrict__ Op, const float* __restrict__ S,
    const float* __restrict__ Rv, const float* __restrict__ rel,
    float* __restrict__ Bm)
{
  const int b    = blockIdx.y;
  const int lane = threadIdx.x & 31;
  const int wave = threadIdx.x >> 5;
  const int half = lane >> 4;
  const int l16  = lane & 15;
  const int rbase = blockIdx.x * 256 + wave * 32;
  if (rbase >= R_) return;                      // wave-uniform (R_ % 32 == 0)

  const float* Sb  = S  + (size_t)b * O_ * R_;
  const float* Rb  = Rv + (size_t)b * O_ * R_;
  const float* Opb = Op + (size_t)b * 500 * 32;

  v8f accS0 = {}, accS1 = {}, accR0 = {}, accR1 = {};

  for (int k0 = 0; k0 < O_; k0 += 4) {
    const int k = k0 + 2 * half;
    const v2f bo = *(const v2f*)(Opb + (size_t)((k0 >> 1) + half) * 32 + l16 * 2);
    {
      const int r = rbase + l16;
      v2f as, ar;
      as.x = __builtin_nontemporal_load(Sb + (size_t)k * R_ + r);
      as.y = __builtin_nontemporal_load(Sb + (size_t)(k + 1) * R_ + r);
      ar.x = __builtin_nontemporal_load(Rb + (size_t)k * R_ + r);
      ar.y = __builtin_nontemporal_load(Rb + (size_t)(k + 1) * R_ + r);
      accS0 = WMMA_F32(as, bo, accS0);
      accR0 = WMMA_F32(ar, bo, accR0);
    }
    {
      const int r = rbase + 16 + l16;
      v2f as, ar;
      as.x = __builtin_nontemporal_load(Sb + (size_t)k * R_ + r);
      as.y = __builtin_nontemporal_load(Sb + (size_t)(k + 1) * R_ + r);
      ar.x = __builtin_nontemporal_load(Rb + (size_t)k * R_ + r);
      ar.y = __builtin_nontemporal_load(Rb + (size_t)(k + 1) * R_ + r);
      accS1 = WMMA_F32(as, bo, accS1);
      accR1 = WMMA_F32(ar, bo, accR1);
    }
  }

  // Bm row: [0..5]=sender, [6..11]=receiver, [12]=relation_info, [13..15]=0.
  const int c = l16;
  #pragma unroll
  for (int t = 0; t < 2; ++t) {
    const v8f aS = t ? accS1 : accS0;
    const v8f aR = t ? accR1 : accR0;
    #pragma unroll
    for (int j = 0; j < 8; ++j) {
      const int r = rbase + t * 16 + j + 8 * half;
      float* row = Bm + ((size_t)b * R_ + r) * BMW;
      if (c < NF) {
        row[c]      = aS[j];
        row[NF + c] = aR[j];
      } else if (c == 12) {
        row[12] = rel[(size_t)b * R_ + r];
      } else if (c >= 13) {
        row[c] = 0.f;
      }
    }
  }
}

// ---------------------------------------------------------------------------
// Kernel 2: relational MLP. 4 waves/WG, 16 rows/wave, wave-private LDS slice
// (stride 164: half-wave rows land in different banks), 10 N-tiles in regs.
// All B loads are unpredicated float2 from pair-interleaved padded weights.
// Output E stored pair-interleaved for kernel 3's B operand.
// ---------------------------------------------------------------------------
__global__ __launch_bounds__(128) void k_relmlp(
    const float* __restrict__ Bm, const float* __restrict__ W1p,
    const float* __restrict__ W2p, const float* __restrict__ W3p,
    const float* __restrict__ W4p, const float* __restrict__ bp,
    float* __restrict__ Ep)
{
  __shared__ float act[64 * 164];
  const int lane = threadIdx.x & 31;
  const int wave = threadIdx.x >> 5;
  const int half = lane >> 4;
  const int l16  = lane & 15;
  const int row0 = blockIdx.x * 64 + wave * 16; // 80000 rows total, exact
  float* my = act + wave * 16 * 164;

  v8f acc[10];

  // ---- layer 1: K=16 (13 real), N=160 (150 real) ----
  #pragma unroll
  for (int n = 0; n < 10; ++n) acc[n] = (v8f){};
  #pragma unroll
  for (int k0 = 0; k0 < 16; k0 += 4) {
    const int k = k0 + 2 * half;
    const v2f a = *(const v2f*)(Bm + (size_t)(row0 + l16) * BMW + k);
    const float* wrow = W1p + (size_t)((k0 >> 1) + half) * 320 + l16 * 2;
    #pragma unroll
    for (int n = 0; n < 10; ++n) {
      const v2f w = *(const v2f*)(wrow + n * 32);
      acc[n] = WMMA_F32(a, w, acc[n]);
    }
  }
  #pragma unroll
  for (int n = 0; n < 10; ++n) {
    const float bias = bp[BO_B1 + n * 16 + l16];
    #pragma unroll
    for (int j = 0; j < 8; ++j) {
      const float v = acc[n][j] + bias;
      my[(j + 8 * half) * 164 + n * 16 + l16] = v > 0.f ? v : 0.f;
    }
  }

  // ---- layers 2 & 3: K=160, N=160 ----
  for (int layer = 0; layer < 2; ++layer) {
    const float* W  = layer ? W3p : W2p;
    const int    bo = layer ? BO_B3 : BO_B2;
    #pragma unroll
    for (int n = 0; n < 10; ++n) acc[n] = (v8f){};
    for (int k0 = 0; k0 < 160; k0 += 4) {
      const int k = k0 + 2 * half;
      const v2f a = *(const v2f*)(my + l16 * 164 + k);
      const float* wrow = W + (size_t)((k0 >> 1) + half) * 320 + l16 * 2;
      #pragma unroll
      for (int n = 0; n < 10; ++n) {
        const v2f w = *(const v2f*)(wrow + n * 32);
        acc[n] = WMMA_F32(a, w, acc[n]);
      }
    }
    #pragma unroll
    for (int n = 0; n < 10; ++n) {
      const float bias = bp[bo + n * 16 + l16];
      #pragma unroll
      for (int j = 0; j < 8; ++j) {
        const float v = acc[n][j] + bias;
        my[(j + 8 * half) * 164 + n * 16 + l16] = v > 0.f ? v : 0.f;
      }
    }
  }

  // ---- layer 4: K=160, N=112 (100 real), ReLU, write pair-interleaved E ----
  v8f a4[7];
  #pragma unroll
  for (int n = 0; n < 7; ++n) a4[n] = (v8f){};
  for (int k0 = 0; k0 < 160; k0 += 4) {
    const int k = k0 + 2 * half;
    const v2f a = *(const v2f*)(my + l16 * 164 + k);
    const float* wrow = W4p + (size_t)((k0 >> 1) + half) * 224 + l16 * 2;
    #pragma unroll
    for (int n = 0; n < 7; ++n) {
      const v2f w = *(const v2f*)(wrow + n * 32);
      a4[n] = WMMA_F32(a, w, a4[n]);
    }
  }
  #pragma unroll
  for (int n = 0; n < 7; ++n) {
    const int nn = n * 16 + l16;
    const float bias = bp[BO_B4 + nn];
    #pragma unroll
    for (int j = 0; j < 8; ++j) {
      const int row = row0 + j + 8 * half;     // global flat row (b*R + r)
      const float v = a4[n][j] + bias;         // nn>=100 -> 0+0
      Ep[(size_t)(row >> 1) * (2 * EW) + nn * 2 + (row & 1)] = v > 0.f ? v : 0.f;
    }
  }
}

// ---------------------------------------------------------------------------
// Kernel 3: aggregation. A[m,k] = Rcv[b,m,k] (NT-streamed, 320 MB once),
// B = pair-interleaved E (L2-resident). 7 N-tiles in regs over K=20000.
// ---------------------------------------------------------------------------
__global__ __launch_bounds__(128) void k_aggregate(
    const float* __restrict__ Rv, const float* __restrict__ Ep,
    const float* __restrict__ objects, float* __restrict__ C)
{
  const int b    = blockIdx.y;
  const int lane = threadIdx.x & 31;
  const int wave = threadIdx.x >> 5;
  const int half = lane >> 4;
  const int l16  = lane & 15;
  const int mt   = blockIdx.x * 4 + wave;
  if (mt >= 63) return;                         // wave-uniform guard

  const int  m    = mt * 16 + l16;
  const bool mval = (m < O_);
  const float* Arow = Rv + ((size_t)b * O_ + (mval ? m : 0)) * R_;
  const float* Eb   = Ep + (size_t)b * (R_ / 2) * (2 * EW);

  v8f acc[7];
  #pragma unroll
  for (int n = 0; n < 7; ++n) acc[n] = (v8f){};

  for (int k0 = 0; k0 < R_; k0 += 4) {
    const int k = k0 + 2 * half;
    const v2f av = __builtin_nontemporal_load((const v2f*)(Arow + k));
    v2f a;
    a.x = mval ? av.x : 0.f;
    a.y = mval ? av.y : 0.f;
    const float* erow = Eb + (size_t)((k0 >> 1) + half) * (2 * EW) + l16 * 2;
    #pragma unroll
    for (int n = 0; n < 7; ++n) {
      const v2f w = *(const v2f*)(erow + n * 32);
      acc[n] = WMMA_F32(a, w, acc[n]);
    }
  }

  #pragma unroll
  for (int n = 0; n < 7; ++n) {
    const int f = n * 16 + l16;                 // effect feature 0..111
    #pragma unroll
    for (int j = 0; j < 8; ++j) {
      const int mm = mt * 16 + j + 8 * half;
      if (mm < O_) {
        float* row = C + ((size_t)b * O_ + mm) * CW;
        if (f < ED) row[NF + f] = acc[n][j];
        if (n == 0 && l16 < NF) {
          row[l16]           = objects[((size_t)b * O_ + mm) * NF + l16];
          row[NF + ED + l16] = 0.f;             // cols 106..111
        }
      }
    }
  }
}

// ---------------------------------------------------------------------------
// Kernel 4: object MLP 106->100->2 over 4000 rows (250 M-tiles).
// ---------------------------------------------------------------------------
__global__ __launch_bounds__(128) void k_objmlp(
    const float* __restrict__ C, const float* __restrict__ OW1p,
    const float* __restrict__ OW2p, const float* __restrict__ bp,
    float* __restrict__ out)
{
  __shared__ float act[4 * 16 * 116];
  const int lane = threadIdx.x & 31;
  const int wave = threadIdx.x >> 5;
  const int half = lane >> 4;
  const int l16  = lane & 15;
  const int mt   = blockIdx.x * 4 + wave;
  if (mt >= 250) return;                        // wave-uniform guard
  float* my = act + wave * 16 * 116;
  const int row0 = mt * 16;

  // ---- layer 1: K=112 (106 real), N=112 (100 real), ReLU ----
  v8f acc[7];
  #pragma unroll
  for (int n = 0; n < 7; ++n) acc[n] = (v8f){};
  for (int k0 = 0; k0 < CW; k0 += 4) {
    const int k = k0 + 2 * half;
    const v2f a = *(const v2f*)(C + (size_t)(row0 + l16) * CW + k);
    const float* wrow = OW1p + (size_t)((k0 >> 1) + half) * 224 + l16 * 2;
    #pragma unroll
    for (int n = 0; n < 7; ++n) {
      const v2f w = *(const v2f*)(wrow + n * 32);
      acc[n] = WMMA_F32(a, w, acc[n]);
    }
  }
  #pragma unroll
  for (int n = 0; n < 7; ++n) {
    const float bias = bp[BO_OB1 + n * 16 + l16];
    #pragma unroll
    for (int j = 0; j < 8; ++j) {
      const float v = acc[n][j] + bias;
      my[(j + 8 * half) * 116 + n * 16 + l16] = v > 0.f ? v : 0.f;
    }
  }

  // ---- layer 2: K=112 (100 real), N=16 (2 real), no ReLU ----
  v8f o2 = {};
  for (int k0 = 0; k0 < 112; k0 += 4) {
    const int k = k0 + 2 * half;
    const v2f a = *(const v2f*)(my + l16 * 116 + k);
    const v2f w = *(const v2f*)(OW2p + (size_t)((k0 >> 1) + half) * 32 + l16 * 2);
    o2 = WMMA_F32(a, w, o2);
  }
  #pragma unroll
  for (int j = 0; j < 8; ++j) {
    const int mm = row0 + j + 8 * half;
    if (l16 < 2) out[(size_t)mm * 2 + l16] = o2[j] + bp[BO_OB2 + l16];
  }
}

// ---------------------------------------------------------------------------
extern "C" void kernel_launch(void* const* d_in, const int* in_sizes, int n_in,
                              void* d_out, int out_size, void* d_ws, size_t ws_size,
                              hipStream_t stream) {
  const float* objects = (const float*)d_in[0];
  const float* S       = (const float*)d_in[1];
  const float* Rv      = (const float*)d_in[2];
  const float* rel     = (const float*)d_in[3];
  const float* rw1 = (const float*)d_in[4];  const float* rb1 = (const float*)d_in[5];
  const float* rw2 = (const float*)d_in[6];  const float* rb2 = (const float*)d_in[7];
  const float* rw3 = (const float*)d_in[8];  const float* rb3 = (const float*)d_in[9];
  const float* rw4 = (const float*)d_in[10]; const float* rb4 = (const float*)d_in[11];
  const float* ow1 = (const float*)d_in[12]; const float* ob1 = (const float*)d_in[13];
  const float* ow2 = (const float*)d_in[14]; const float* ob2 = (const float*)d_in[15];
  float* out = (float*)d_out;

  float* ws = (float*)d_ws;
  float* Bm = ws + OFF_BM;
  float* Ep = ws + OFF_EP;
  float* C  = ws + OFF_C;

  k_prep<<<dim3(64), 256, 0, stream>>>(objects, rw1, rb1, rw2, rb2, rw3, rb3,
                                       rw4, rb4, ow1, ob1, ow2, ob2, ws);

  dim3 g1((R_ + 255) / 256, B_);
  k_marshal<<<g1, 256, 0, stream>>>(ws + OFF_OP, S, Rv, rel, Bm);

  k_relmlp<<<dim3((B_ * R_) / 64), 128, 0, stream>>>(
      Bm, ws + OFF_W1, ws + OFF_W2, ws + OFF_W3, ws + OFF_W4, ws + OFF_BIAS, Ep);

  dim3 g3(16, B_);
  k_aggregate<<<g3, 128, 0, stream>>>(Rv, Ep, objects, C);

  k_objmlp<<<dim3(63), 128, 0, stream>>>(C, ws + OFF_OW1, ws + OFF_OW2,
                                         ws + OFF_BIAS, out);
}